// VQVAE_6227702580026
// MI455X (gfx1250) — compile-verified
//
#include <hip/hip_runtime.h>
#include <hip/hip_bf16.h>
#include <cstdint>
#include <cstddef>

typedef float v2f __attribute__((ext_vector_type(2)));
typedef float v8f __attribute__((ext_vector_type(8)));

// ---------------------------------------------------------------------------
// conv1: Cin=1, K=4, stride 2, pad 1.  x:(B,Lin) -> y:(Cout,B,Lout) + bias
// ---------------------------------------------------------------------------
__global__ void conv1_kernel(const float* __restrict__ x, const float* __restrict__ w,
                             const float* __restrict__ bias, float* __restrict__ y,
                             int Bn, int Lin, int Lout, int Cout) {
  int i = blockIdx.x * blockDim.x + threadIdx.x;
  int total = Cout * Bn * Lout;
  if (i >= total) return;
  int c = i / (Bn * Lout);
  int rem = i - c * (Bn * Lout);
  int b = rem / Lout;
  int l = rem - b * Lout;
  float acc = bias[c];
#pragma unroll
  for (int k = 0; k < 4; ++k) {
    int s = 2 * l + k - 1;
    if (s >= 0 && s < Lin) acc += w[c * 4 + k] * x[(size_t)b * Lin + s];
  }
  y[i] = acc;
}

// ---------------------------------------------------------------------------
// BatchNorm (training stats over B*L per channel) -> scale/shift
// ---------------------------------------------------------------------------
__global__ void bn_stats_kernel(const float* __restrict__ data, const float* __restrict__ g,
                                const float* __restrict__ beta, float* __restrict__ scale,
                                float* __restrict__ shift, int Nl) {
  __shared__ float ss[256];
  __shared__ float sq[256];
  int c = blockIdx.x;
  const float* p = data + (size_t)c * Nl;
  float s = 0.f, q = 0.f;
  for (int i = threadIdx.x; i < Nl; i += 256) { float v = p[i]; s += v; q += v * v; }
  ss[threadIdx.x] = s; sq[threadIdx.x] = q;
  __syncthreads();
  for (int o = 128; o > 0; o >>= 1) {
    if (threadIdx.x < o) { ss[threadIdx.x] += ss[threadIdx.x + o]; sq[threadIdx.x] += sq[threadIdx.x + o]; }
    __syncthreads();
  }
  if (threadIdx.x == 0) {
    float inv = 1.0f / (float)Nl;
    float m = ss[0] * inv;
    float v = sq[0] * inv - m * m;           // biased variance
    float sc = g[c] * rsqrtf(v + 1e-5f);
    scale[c] = sc;
    shift[c] = beta[c] - m * sc;
  }
}

__global__ void bn_apply_kernel(float* __restrict__ data, const float* __restrict__ scale,
                                const float* __restrict__ shift, int Nl, int total) {
  int i = blockIdx.x * blockDim.x + threadIdx.x;
  if (i >= total) return;
  int c = i / Nl;
  float v = data[i] * scale[c] + shift[c];
  data[i] = v > 0.f ? v : 0.f;
}

// ---------------------------------------------------------------------------
// WMMA f32 GEMM:  C[M][N] = A(MxK) * B(KxN) (+bias[m])
// MODE 0: plain          A[m*lda+k],                    B[k*ldb + n0 + n]
// MODE 1: conv  s2 p1 k4 A[m*lda+k] (Cout,Cin*4),       B = im2col of X:(Cin,Bn,Lin)
// MODE 2: convT s2 p1 k4 A = W:(Cin,Cout,4) gathered,   B = frac-strided gather of X
//
// Block: 256 threads (8 wave32), 64(M) x 128(N) tile, K-chunk 16 staged in LDS.
// LDS holds k-PAIR interleaved float2 so every WMMA fragment is one ds_load_b64:
//   A pair-pitch 9  (word stride 18: conflict-free, halves never collide)
//   B pair-pitch 144 (2*144 = 32 mod 64: hi/lo lane halves in disjoint banks)
// Each wave owns 1 M-subtile x 4 N-subtiles -> 16 v_wmma per K-chunk.
// M % 64 == 0, N % 128 == 0, K % 16 == 0 (true at every call site).
// ---------------------------------------------------------------------------
template <int MODE>
__global__ void __launch_bounds__(256) gemm_wmma_kernel(
    const float* __restrict__ A, const float* __restrict__ Bsrc, float* __restrict__ Cd,
    const float* __restrict__ bias, int M, int Kd, int lda, int ldb, int ldc, int n0,
    int Bn, int Lin, int Lout) {
  __shared__ v2f AsP[64 * 9];     // [m][kpair]
  __shared__ v2f BsP[8 * 144];    // [kpair][n]
  float* Asf = (float*)AsP;
  float* Bsf = (float*)BsP;

  const int tid = threadIdx.x;
  const int m0 = blockIdx.y * 64;
  const int ntile = blockIdx.x * 128;
  const int wave = tid >> 5, lane = tid & 31;
  const int mi = wave >> 1;        // M-subtile 0..3
  const int nj = wave & 1;         // N-subtile quad 0..1
  const int lm = lane & 15, hi = lane >> 4;
  const int NBLin = Bn * Lin;

  const v8f vzero = {0.f, 0.f, 0.f, 0.f, 0.f, 0.f, 0.f, 0.f};
  v8f acc[4] = {vzero, vzero, vzero, vzero};

  for (int kc = 0; kc < Kd; kc += 16) {
    __syncthreads();
    // ---- cooperative A tile: 64 rows x 16 k ----
#pragma unroll
    for (int i = 0; i < 4; ++i) {
      int e = tid + i * 256;
      int mr = e >> 4, kk = e & 15;
      int kg = kc + kk;
      float v;
      if (MODE == 2) {
        int ci = kg >> 2, r = kg & 3;
        v = A[(size_t)ci * (M * 4) + (size_t)(m0 + mr) * 4 + r];
      } else {
        v = A[(size_t)(m0 + mr) * lda + kg];
      }
      Asf[mr * 18 + kk] = v;       // pair (k,k+1) lands contiguously
    }
    // ---- cooperative B tile: 16 k x 128 n (fused im2col / frac-stride) ----
#pragma unroll
    for (int i = 0; i < 8; ++i) {
      int e = tid + i * 256;
      int kk = e >> 7, nn = e & 127;
      int kg = kc + kk;
      int ng = ntile + nn;
      float v;
      if (MODE == 0) {
        v = Bsrc[(size_t)kg * ldb + n0 + ng];
      } else {
        int ci = kg >> 2, r = kg & 3;
        int b = ng / Lout;
        int l = ng - b * Lout;
        int s; bool ok;
        if (MODE == 1) {                 // conv: s = 2l + r - 1
          s = 2 * l + r - 1;
          ok = (s >= 0) && (s < Lin);
        } else {                          // convT: t = 2s + r - 1
          int num = l + 1 - r;
          s = num >> 1;
          ok = (num >= 0) && ((num & 1) == 0) && (s < Lin);
        }
        v = ok ? Bsrc[(size_t)ci * NBLin + (size_t)b * Lin + s] : 0.0f;
      }
      Bsf[((kk >> 1) * 144 + nn) * 2 + (kk & 1)] = v;
    }
    __syncthreads();
    // ---- 4 k-steps x 4 N-subtiles of V_WMMA_F32_16X16X4_F32 ----
#pragma unroll
    for (int ks = 0; ks < 4; ++ks) {
      // lane layout: lo half holds k0,k0+1; hi half holds k0+2,k0+3
      v2f a = AsP[(mi * 16 + lm) * 9 + ks * 2 + hi];
#pragma unroll
      for (int s = 0; s < 4; ++s) {
        v2f bb = BsP[(ks * 2 + hi) * 144 + (nj * 4 + s) * 16 + lm];
        acc[s] = __builtin_amdgcn_wmma_f32_16x16x4_f32(false, a, false, bb, (short)0,
                                                       acc[s], false, false);
      }
    }
  }
  // ---- epilogue: C/D layout -> VGPR r holds M = r + 8*hi ----
#pragma unroll
  for (int s = 0; s < 4; ++s) {
#pragma unroll
    for (int r = 0; r < 8; ++r) {
      int m = m0 + mi * 16 + r + hi * 8;
      int n = ntile + (nj * 4 + s) * 16 + lm;
      float v = acc[s][r];
      if (bias) v += bias[m];
      Cd[(size_t)m * ldc + n] = v;
    }
  }
}

// ---------------------------------------------------------------------------
// VQ helpers
// ---------------------------------------------------------------------------
__global__ void vq_init_kernel(float* __restrict__ counts, float* __restrict__ vqsum, int K) {
  int i = blockIdx.x * blockDim.x + threadIdx.x;
  if (i < K) counts[i] = 0.f;
  if (i == 0) vqsum[0] = 0.f;
}

__global__ void cnorm_kernel(const float* __restrict__ cb, float* __restrict__ cn, int K, int D) {
  int j = blockIdx.x * blockDim.x + threadIdx.x;
  if (j >= K) return;
  float s = 0.f;
  for (int d = 0; d < D; ++d) { float v = cb[(size_t)j * D + d]; s += v * v; }
  cn[j] = s;
}

// dist(n,j) ~ cnorm[j] - 2*G[j][n]  (||z||^2 constant in j -> drop for argmin)
__global__ void argmin_kernel(const float* __restrict__ G, const float* __restrict__ cn,
                              int* __restrict__ idx, float* __restrict__ counts,
                              int n0, int CH, int Kc) {
  int n = blockIdx.x * blockDim.x + threadIdx.x;
  if (n >= CH) return;
  float best = 3.4e38f; int bj = 0;
  for (int j = 0; j < Kc; ++j) {
    float d = cn[j] - 2.0f * G[(size_t)j * CH + n];
    if (d < best) { best = d; bj = j; }    // first minimum, like jnp.argmin
  }
  idx[n0 + n] = bj;
  atomicAdd(&counts[bj], 1.0f);
}

// zq = codebook[idx] (straight-through forward value); accumulate sum (q-z)^2
__global__ void gather_loss_kernel(const float* __restrict__ cb, const int* __restrict__ idx,
                                   const float* __restrict__ z, float* __restrict__ zq,
                                   float* __restrict__ vqsum, int N, int D) {
  __shared__ float red[256];
  long long i = (long long)blockIdx.x * blockDim.x + threadIdx.x;
  float contrib = 0.f;
  if (i < (long long)N * D) {
    int d = (int)(i / N);
    int n = (int)(i - (long long)d * N);
    float q = cb[(size_t)idx[n] * D + d];
    float zv = z[i];
    zq[i] = q;
    float df = q - zv;
    contrib = df * df;
  }
  red[threadIdx.x] = contrib;
  __syncthreads();
  for (int o = 128; o > 0; o >>= 1) {
    if (threadIdx.x < o) red[threadIdx.x] += red[threadIdx.x + o];
    __syncthreads();
  }
  if (threadIdx.x == 0) atomicAdd(vqsum, red[0]);
}

// ---------------------------------------------------------------------------
// final 1x1 convT head: (64,B,L) -> (B*L) with 64-ch dot
// ---------------------------------------------------------------------------
__global__ void final_conv_kernel(const float* __restrict__ x, const float* __restrict__ w,
                                  const float* __restrict__ bias, float* __restrict__ out,
                                  int Cin, int NBL) {
  int n = blockIdx.x * blockDim.x + threadIdx.x;
  if (n >= NBL) return;
  float s = bias[0];
  for (int c = 0; c < Cin; ++c) s += w[c] * x[(size_t)c * NBL + n];
  out[n] = s;
}

// vq_loss + perplexity scalars
__global__ void finalize_kernel(const float* __restrict__ counts, const float* __restrict__ vqsum,
                                float* __restrict__ out2, int K, float Nf, float loss_scale) {
  __shared__ float red[512];
  int j = threadIdx.x;
  float p = (j < K) ? counts[j] / Nf : 0.f;
  red[j] = -p * logf(p + 1e-10f);
  __syncthreads();
  for (int o = 256; o > 0; o >>= 1) {
    if (j < o) red[j] += red[j + o];
    __syncthreads();
  }
  if (j == 0) {
    out2[0] = loss_scale * vqsum[0];   // (1 + 0.25) * mean((q-z)^2)
    out2[1] = expf(red[0]);
  }
}

__global__ void idx_out_kernel(const int* __restrict__ idx, float* __restrict__ out, int N) {
  int i = blockIdx.x * blockDim.x + threadIdx.x;
  if (i < N) out[i] = (float)idx[i];
}

// ---------------------------------------------------------------------------
// Host launcher
// ---------------------------------------------------------------------------
extern "C" void kernel_launch(void* const* d_in, const int* in_sizes, int n_in,
                              void* d_out, int out_size, void* d_ws, size_t ws_size,
                              hipStream_t stream) {
  (void)in_sizes; (void)n_in; (void)out_size; (void)ws_size;
  const float* x   = (const float*)d_in[0];
  const float* ew[4] = {(const float*)d_in[1], (const float*)d_in[2],
                        (const float*)d_in[3], (const float*)d_in[4]};
  const float* eb[4] = {(const float*)d_in[5], (const float*)d_in[6],
                        (const float*)d_in[7], (const float*)d_in[8]};
  const float* eg[3] = {(const float*)d_in[9], (const float*)d_in[10], (const float*)d_in[11]};
  const float* ebt[3] = {(const float*)d_in[12], (const float*)d_in[13], (const float*)d_in[14]};
  const float* dw[4] = {(const float*)d_in[15], (const float*)d_in[16],
                        (const float*)d_in[17], (const float*)d_in[18]};
  const float* db[4] = {(const float*)d_in[19], (const float*)d_in[20],
                        (const float*)d_in[21], (const float*)d_in[22]};
  const float* dg[3] = {(const float*)d_in[23], (const float*)d_in[24], (const float*)d_in[25]};
  const float* dbt[3] = {(const float*)d_in[26], (const float*)d_in[27], (const float*)d_in[28]};
  const float* cb = (const float*)d_in[29];

  // workspace layout (floats); needs ~143MB
  float* ws = (float*)d_ws;
  float* buf0  = ws;                         // 16,777,216
  float* buf1  = ws + 16777216;              // 16,777,216
  float* G     = ws + 33554432;              // 512*4096 VQ distance tile
  float* scale = ws + 35651584;              // 512
  float* shift = scale + 512;                // 512
  float* cn    = shift + 512;                // 512
  float* counts = cn + 512;                  // 512
  float* vqsum  = counts + 512;              // 1 (+pad)
  int*   idxb   = (int*)(vqsum + 32);        // 32768 ints

  const int Bn = 64;
  float* out = (float*)d_out;

  // ----- encoder -----
  conv1_kernel<<<(64 * Bn * 2048 + 255) / 256, 256, 0, stream>>>(x, ew[0], eb[0], buf0, Bn, 4096, 2048, 64);
  bn_stats_kernel<<<64, 256, 0, stream>>>(buf0, eg[0], ebt[0], scale, shift, Bn * 2048);
  bn_apply_kernel<<<(64 * Bn * 2048 + 255) / 256, 256, 0, stream>>>(buf0, scale, shift, Bn * 2048, 64 * Bn * 2048);

  gemm_wmma_kernel<1><<<dim3(65536 / 128, 128 / 64), 256, 0, stream>>>(
      ew[1], buf0, buf1, eb[1], 128, 256, 256, 0, 65536, 0, Bn, 2048, 1024);
  bn_stats_kernel<<<128, 256, 0, stream>>>(buf1, eg[1], ebt[1], scale, shift, 65536);
  bn_apply_kernel<<<(128 * 65536 + 255) / 256, 256, 0, stream>>>(buf1, scale, shift, 65536, 128 * 65536);

  gemm_wmma_kernel<1><<<dim3(32768 / 128, 256 / 64), 256, 0, stream>>>(
      ew[2], buf1, buf0, eb[2], 256, 512, 512, 0, 32768, 0, Bn, 1024, 512);
  bn_stats_kernel<<<256, 256, 0, stream>>>(buf0, eg[2], ebt[2], scale, shift, 32768);
  bn_apply_kernel<<<(256 * 32768 + 255) / 256, 256, 0, stream>>>(buf0, scale, shift, 32768, 256 * 32768);

  // conv4 1x1 (plain GEMM): z = W(512x256) * e3(256x32768) -> buf1 (D,N)
  gemm_wmma_kernel<0><<<dim3(32768 / 128, 512 / 64), 256, 0, stream>>>(
      ew[3], buf0, buf1, eb[3], 512, 256, 256, 32768, 32768, 0, 0, 0, 0);

  // ----- vector quantization (chunked over N) -----
  vq_init_kernel<<<2, 256, 0, stream>>>(counts, vqsum, 512);
  cnorm_kernel<<<2, 256, 0, stream>>>(cb, cn, 512, 512);
  for (int ch = 0; ch < 8; ++ch) {
    gemm_wmma_kernel<0><<<dim3(4096 / 128, 512 / 64), 256, 0, stream>>>(
        cb, buf1, G, nullptr, 512, 512, 512, 32768, 4096, ch * 4096, 0, 0, 0);
    argmin_kernel<<<16, 256, 0, stream>>>(G, cn, idxb, counts, ch * 4096, 4096, 512);
  }
  // zq = codebook[idx] -> buf0; accumulate (q-z)^2
  gather_loss_kernel<<<65536, 256, 0, stream>>>(cb, idxb, buf1, buf0, vqsum, 32768, 512);

  // ----- decoder -----
  gemm_wmma_kernel<2><<<dim3(65536 / 128, 256 / 64), 256, 0, stream>>>(
      dw[0], buf0, buf1, db[0], 256, 2048, 0, 0, 65536, 0, Bn, 512, 1024);
  bn_stats_kernel<<<256, 256, 0, stream>>>(buf1, dg[0], dbt[0], scale, shift, 65536);
  bn_apply_kernel<<<(256 * 65536 + 255) / 256, 256, 0, stream>>>(buf1, scale, shift, 65536, 256 * 65536);

  gemm_wmma_kernel<2><<<dim3(131072 / 128, 128 / 64), 256, 0, stream>>>(
      dw[1], buf1, buf0, db[1], 128, 1024, 0, 0, 131072, 0, Bn, 1024, 2048);
  bn_stats_kernel<<<128, 256, 0, stream>>>(buf0, dg[1], dbt[1], scale, shift, 131072);
  bn_apply_kernel<<<(128 * 131072 + 255) / 256, 256, 0, stream>>>(buf0, scale, shift, 131072, 128 * 131072);

  gemm_wmma_kernel<2><<<dim3(262144 / 128, 64 / 64), 256, 0, stream>>>(
      dw[2], buf0, buf1, db[2], 64, 512, 0, 0, 262144, 0, Bn, 2048, 4096);
  bn_stats_kernel<<<64, 256, 0, stream>>>(buf1, dg[2], dbt[2], scale, shift, 262144);
  bn_apply_kernel<<<(64 * 262144 + 255) / 256, 256, 0, stream>>>(buf1, scale, shift, 262144, 64 * 262144);

  final_conv_kernel<<<(262144 + 255) / 256, 256, 0, stream>>>(buf1, dw[3], db[3], out, 64, 262144);

  // ----- scalars + indices -----
  finalize_kernel<<<1, 512, 0, stream>>>(counts, vqsum, out + 262144, 512,
                                         32768.0f, 1.25f / (32768.0f * 512.0f));
  idx_out_kernel<<<(32768 + 255) / 256, 256, 0, stream>>>(idxb, out + 262146, 32768);
}